// Generative_17669495456168
// MI455X (gfx1250) — compile-verified
//
#include <hip/hip_runtime.h>
#include <hip/hip_bf16.h>
#include <math.h>

// ---------------------------------------------------------------------------
// Task-conditional VAE forward, MI455X (gfx1250), wave32, fp32 WMMA 16x16x4.
// Pipeline: bucket rows by task -> segmented WMMA GEMMs -> reparam -> decoder
// GEMMs -> per-task sigmoid head with scatter.
// ---------------------------------------------------------------------------

typedef float v2f __attribute__((ext_vector_type(2)));
typedef float v8f __attribute__((ext_vector_type(8)));

#define BQ 8192   // batch
#define DQ 784    // input dim
#define HQ 1024   // hidden
#define LQ 128    // latent
#define TQ 10     // tasks

#define BM 32
#define BN 128
#define BK 32

#define LDA_STRIDE 36                 // 32 + pad, multiple of 4 (16B rows)
#define LDB_STRIDE 257                // 2*BN + 1 (odd -> hi half-wave on odd banks)

// ---------------------------- bookkeeping kernels ---------------------------

__global__ void k_init(int* counts, int* dec_offs) {
    int i = threadIdx.x;
    if (i < TQ) counts[i] = 0;
    if (i == 0) { dec_offs[0] = 0; dec_offs[1] = BQ; }
}

__global__ void k_bucket(const int* __restrict__ task, int* counts,
                         int* __restrict__ perm_raw) {
    int i = blockIdx.x * blockDim.x + threadIdx.x;
    if (i < BQ) {
        int t = task[i];
        int pos = atomicAdd(&counts[t], 1);
        perm_raw[t * BQ + pos] = i;
    }
}

__global__ void k_scan(const int* __restrict__ counts, int* __restrict__ offs) {
    if (threadIdx.x == 0) {
        int s = 0;
        for (int t = 0; t < TQ; ++t) { offs[t] = s; s += counts[t]; }
        offs[TQ] = s;
    }
}

__global__ void k_compact(const int* __restrict__ counts, const int* __restrict__ offs,
                          const int* __restrict__ perm_raw, int* __restrict__ cperm) {
    int t = blockIdx.y;
    int j = blockIdx.x * blockDim.x + threadIdx.x;
    if (j < counts[t]) cperm[offs[t] + j] = perm_raw[t * BQ + j];
}

// z = mu + exp(ls)*eps ; also scatter mu / log_sigma into d_out
__global__ void k_reparam(const float* __restrict__ encout, const float* __restrict__ eps,
                          const int* __restrict__ cperm, float* __restrict__ z,
                          float* __restrict__ out_mu, float* __restrict__ out_ls) {
    int idx = blockIdx.x * blockDim.x + threadIdx.x;
    if (idx >= BQ * LQ) return;
    int gp = idx / LQ, l = idx % LQ;
    int r = cperm[gp];
    float mu = encout[gp * (2 * LQ) + l];
    float ls = encout[gp * (2 * LQ) + LQ + l];
    out_mu[r * LQ + l] = mu;
    out_ls[r * LQ + l] = ls;
    z[gp * LQ + l] = mu + expf(ls) * eps[r * LQ + l];
}

// ------------------------------- WMMA GEMM ----------------------------------
// C[seg rows] = act( A @ W[t] + bias[t] )
// EPI: 0 = relu, store grouped   1 = identity, store grouped
//      2 = sigmoid, scatter rows via cperm
template <int EPI>
__global__ __launch_bounds__(256) void gemm_seg(
    const float* __restrict__ A, const float* __restrict__ W,
    const float* __restrict__ bias, float* __restrict__ Cout,
    const int* __restrict__ cperm, const int* __restrict__ offs,
    int K, int N, int gatherA, unsigned long long wstride) {

    __shared__ float ldsA[BM][LDA_STRIDE];
    __shared__ float ldsBp[(BK / 2) * LDB_STRIDE];   // pair-packed [kpair][n*2+pos]

    const int t = blockIdx.z;
    const int seg = offs[t];
    const int cnt = offs[t + 1] - seg;
    const int mtile = blockIdx.y * BM;
    if (mtile >= cnt) return;                        // uniform exit
    const int nbase = blockIdx.x * BN;

    const float* Wt = W + (unsigned long long)t * wstride;
    const float* bt = bias + (unsigned long long)t * N;

    const int tid = threadIdx.x;
    const int lane = tid & 31;
    const int wave = tid >> 5;
    const int wr = wave & 1;        // row half (16 rows)
    const int wc = wave >> 1;       // col quarter (32 cols)
    const int lane15 = lane & 15;
    const int laneHi = lane >> 4;

    // ---- hoisted A staging geometry: one (row, k-quad) per thread ----
    const int am = tid >> 3;                 // 0..31 row in tile
    const int ak = (tid & 7) * 4;            // 0..28 k-quad
    int mc = mtile + am; if (mc > cnt - 1) mc = cnt - 1;   // clamp partial tile
    int gpa = seg + mc;
    int ra = gatherA ? cperm[gpa] : gpa;
    const float* arow = A + (unsigned long long)ra * K;

    v8f acc0 = {};
    v8f acc1 = {};

    for (int kk = 0; kk < K; kk += BK) {
        // ---- stage A tile (32 x 32): one float4 + b128 LDS store ----
        {
            int kg = kk + ak;                         // K % 4 == 0 always
            float4 v = make_float4(0.f, 0.f, 0.f, 0.f);
            if (kg < K) v = *(const float4*)(arow + kg);
            *(float4*)&ldsA[am][ak] = v;
        }
        // ---- stage B tile (32 x 128): 8 float2, pair-packed stores ----
#pragma unroll
        for (int i = 0; i < 8; ++i) {
            int e = tid + i * 256;                    // 0..2047
            int bk = e >> 6;                          // 0..31
            int bn = (e & 63) * 2;                    // 0..126 step 2
            int kg = kk + bk, ng = nbase + bn;
            float2 v = make_float2(0.f, 0.f);
            if (kg < K && ng < N) v = *(const float2*)(Wt + (unsigned long long)kg * N + ng);
            float* p = &ldsBp[(bk >> 1) * LDB_STRIDE + bn * 2 + (bk & 1)];
            p[0] = v.x;                               // (k, n)
            p[2] = v.y;                               // (k, n+1)
        }
        __syncthreads();

        // ---- 8 k-steps of V_WMMA_F32_16X16X4_F32, two N-tiles per wave ----
#pragma unroll
        for (int s = 0; s < 8; ++s) {
            int kl = s * 4 + laneHi * 2;              // lanes<16:K0..1, lanes>=16:K2..3
            const float* pa = &ldsA[wr * 16 + lane15][kl];
            v2f a; a.x = pa[0]; a.y = pa[1];
            const float* pb0 = &ldsBp[(2 * s + laneHi) * LDB_STRIDE +
                                      (wc * 32 + lane15) * 2];
            v2f b0; b0.x = pb0[0]; b0.y = pb0[1];
            const float* pb1 = pb0 + 32;              // +16 columns
            v2f b1; b1.x = pb1[0]; b1.y = pb1[1];
            acc0 = __builtin_amdgcn_wmma_f32_16x16x4_f32(false, a, false, b0,
                                                         (short)0, acc0, false, false);
            acc1 = __builtin_amdgcn_wmma_f32_16x16x4_f32(false, a, false, b1,
                                                         (short)0, acc1, false, false);
        }
        __syncthreads();
    }

    // ------------------------------ epilogue --------------------------------
    int col0 = nbase + wc * 32 + lane15;
    int col1 = col0 + 16;
    float b0v = (col0 < N) ? bt[col0] : 0.f;
    float b1v = (col1 < N) ? bt[col1] : 0.f;

#pragma unroll
    for (int v = 0; v < 8; ++v) {
        int mlocal = wr * 16 + laneHi * 8 + v;        // C layout: VGPR v -> rows v / v+8
        if (mtile + mlocal < cnt) {
            int gp = seg + mtile + mlocal;
            float x0 = acc0[v] + b0v;
            float x1 = acc1[v] + b1v;
            if (EPI == 0) { x0 = fmaxf(x0, 0.f); x1 = fmaxf(x1, 0.f); }
            if (EPI == 2) {
                x0 = 1.f / (1.f + expf(-x0));
                x1 = 1.f / (1.f + expf(-x1));
                int r = cperm[gp];
                if (col0 < N) Cout[(unsigned long long)r * N + col0] = x0;
                if (col1 < N) Cout[(unsigned long long)r * N + col1] = x1;
            } else {
                if (col0 < N) Cout[(unsigned long long)gp * N + col0] = x0;
                if (col1 < N) Cout[(unsigned long long)gp * N + col1] = x1;
            }
        }
    }
}

// ------------------------------- launcher -----------------------------------

extern "C" void kernel_launch(void* const* d_in, const int* in_sizes, int n_in,
                              void* d_out, int out_size, void* d_ws, size_t ws_size,
                              hipStream_t stream) {
    const float* x      = (const float*)d_in[0];
    const float* eps    = (const float*)d_in[1];
    const float* enc_W1 = (const float*)d_in[2];
    const float* enc_b1 = (const float*)d_in[3];
    const float* enc_W2 = (const float*)d_in[4];
    const float* enc_b2 = (const float*)d_in[5];
    const float* enc_W3 = (const float*)d_in[6];
    const float* enc_b3 = (const float*)d_in[7];
    const float* dec_W1 = (const float*)d_in[8];
    const float* dec_b1 = (const float*)d_in[9];
    const float* dec_W2 = (const float*)d_in[10];
    const float* dec_b2 = (const float*)d_in[11];
    const float* head_W = (const float*)d_in[12];
    const float* head_b = (const float*)d_in[13];
    const int*   task   = (const int*)d_in[14];

    float* out = (float*)d_out;
    float* out_recon = out;                         // [B, D]
    float* out_mu    = out + (size_t)BQ * DQ;       // [B, L]
    float* out_ls    = out_mu + (size_t)BQ * LQ;    // [B, L]

    // workspace layout
    int* counts   = (int*)d_ws;                     // [T]   (pad 16)
    int* offs     = counts + 16;                    // [T+1] (pad 16)
    int* dec_offs = offs + 16;                      // [2]   (pad 16)
    int* perm_raw = dec_offs + 16;                  // [T*B]
    int* cperm    = perm_raw + (size_t)TQ * BQ;     // [B]
    float* bufA   = (float*)(cperm + BQ);           // [B*H]
    float* bufB   = bufA + (size_t)BQ * HQ;         // [B*H]
    float* encout = bufA;                           // [B, 2L]  (aliases bufA head)
    float* zbuf   = bufA + (size_t)BQ * 2 * LQ;     // [B, L]   (after encout)

    const int mtiles = (BQ + BM - 1) / BM;          // 256

    // 1) bucket rows by task
    k_init<<<1, 32, 0, stream>>>(counts, dec_offs);
    k_bucket<<<BQ / 256, 256, 0, stream>>>(task, counts, perm_raw);
    k_scan<<<1, 1, 0, stream>>>(counts, offs);
    k_compact<<<dim3(BQ / 256, TQ), 256, 0, stream>>>(counts, offs, perm_raw, cperm);

    // 2) encoder (per-task segments), rows gathered via cperm for stage 1
    gemm_seg<0><<<dim3(HQ / BN, mtiles, TQ), 256, 0, stream>>>(
        x, enc_W1, enc_b1, bufA, cperm, offs, DQ, HQ, 1,
        (unsigned long long)DQ * HQ);
    gemm_seg<0><<<dim3(HQ / BN, mtiles, TQ), 256, 0, stream>>>(
        bufA, enc_W2, enc_b2, bufB, cperm, offs, HQ, HQ, 0,
        (unsigned long long)HQ * HQ);
    gemm_seg<1><<<dim3((2 * LQ) / BN, mtiles, TQ), 256, 0, stream>>>(
        bufB, enc_W3, enc_b3, encout, cperm, offs, HQ, 2 * LQ, 0,
        (unsigned long long)HQ * 2 * LQ);

    // 3) reparameterization (writes mu / log_sigma to d_out, z grouped)
    k_reparam<<<(BQ * LQ) / 256, 256, 0, stream>>>(encout, eps, cperm, zbuf,
                                                   out_mu, out_ls);

    // 4) shared decoder (single segment covering all rows)
    gemm_seg<0><<<dim3(HQ / BN, mtiles, 1), 256, 0, stream>>>(
        zbuf, dec_W1, dec_b1, bufB, cperm, dec_offs, LQ, HQ, 0, 0ULL);
    gemm_seg<0><<<dim3(HQ / BN, mtiles, 1), 256, 0, stream>>>(
        bufB, dec_W2, dec_b2, bufA, cperm, dec_offs, HQ, HQ, 0, 0ULL);

    // 5) per-task sigmoid head, scatter back to original row order
    gemm_seg<2><<<dim3((DQ + BN - 1) / BN, mtiles, TQ), 256, 0, stream>>>(
        bufA, head_W, head_b, out_recon, cperm, offs, HQ, DQ, 0,
        (unsigned long long)HQ * DQ);
}